// SpGraphAttentionLayer_85847806313255
// MI455X (gfx1250) — compile-verified
//
#include <hip/hip_runtime.h>

#define NN 100000
#define NE 3200000
#define IN_F 256
#define OUT_F 64
#define ALPHA_NEG 0.2f

typedef __attribute__((ext_vector_type(2))) float v2f;
typedef __attribute__((ext_vector_type(8))) float v8f;

// Hardware fp32 atomic add (no-return form, tracked on STOREcnt) — avoids any
// CAS-loop fallback the generic atomicAdd might lower to.
__device__ __forceinline__ void global_atomic_fadd(float* p, float v) {
    asm volatile("global_atomic_add_f32 %0, %1, off" :: "v"(p), "v"(v) : "memory");
}

// ---------------------------------------------------------------------------
// Kernel 0: zero accumulators (d_out holds h_prime accumulation; rowsum in ws)
// ---------------------------------------------------------------------------
__global__ void __launch_bounds__(256) zero_kernel(float* __restrict__ out,
                                                   float* __restrict__ rowsum) {
    int idx = blockIdx.x * 256 + threadIdx.x;
    if (idx < NN * OUT_F) out[idx] = 0.0f;
    if (idx < NN)         rowsum[idx] = 0.0f;
}

// ---------------------------------------------------------------------------
// Kernel 1: h = input @ W via V_WMMA_F32_16X16X4_F32.
// Block = 256 threads = 8 waves; each wave owns a 16-row stripe (128 rows/blk)
// and all 4 N-tiles (OUT_F = 64 = 4 x 16).
// W staged in LDS *K-pair interleaved*: Wl[(k/2)*64 + col] = {W[k][col],
// W[k+1][col]} so each B fragment is one contiguous ds_load_b64 into an
// even-aligned VGPR pair (no v_mov repacking before the WMMA).
// A frag: lane(0-15)->M=lane, K pair {k,k+1}; lane(16-31)->same M, {k+2,k+3}
// ---------------------------------------------------------------------------
__global__ void __launch_bounds__(256)
gemm_h_kernel(const float* __restrict__ input, const float* __restrict__ W,
              float* __restrict__ h) {
    __shared__ v2f Wlds[(IN_F / 2) * OUT_F];  // 64 KB of the 320 KB WGP LDS
    for (int i = threadIdx.x; i < (IN_F / 2) * OUT_F; i += 256) {
        int kk = i >> 6;           // K pair index 0..127
        int col = i & 63;
        v2f w;
        w.x = W[(2 * kk)     * OUT_F + col];
        w.y = W[(2 * kk + 1) * OUT_F + col];
        Wlds[i] = w;
    }
    __syncthreads();

    const int lane = threadIdx.x & 31;
    const int wave = threadIdx.x >> 5;
    const int lo   = lane & 15;
    const int hi   = lane >> 4;
    const int row_base = blockIdx.x * 128 + wave * 16;

    int arow = row_base + lo;
    if (arow >= NN) arow = NN - 1;                 // clamp; stores are guarded
    const float* aptr = input + (size_t)arow * IN_F + 2 * hi;

    v8f acc[4];
    acc[0] = v8f{}; acc[1] = v8f{}; acc[2] = v8f{}; acc[3] = v8f{};

#pragma unroll 4
    for (int k = 0; k < IN_F; k += 4) {
        v2f a = *(const v2f*)(aptr + k);           // contiguous K pair per lane
        const v2f* wrow = Wlds + ((k >> 1) + hi) * OUT_F;
#pragma unroll
        for (int t = 0; t < 4; ++t) {
            v2f b = wrow[t * 16 + lo];             // single ds_load_b64
            acc[t] = __builtin_amdgcn_wmma_f32_16x16x4_f32(
                false, a, false, b, (short)0, acc[t], false, false);
        }
    }

    // C layout: VGPR i -> M = i + 8*hi, N = (lane&15) + 16*t
#pragma unroll
    for (int i = 0; i < 8; ++i) {
        int row = row_base + 8 * hi + i;
        if (row < NN) {
            float* hrow = h + (size_t)row * OUT_F;
#pragma unroll
            for (int t = 0; t < 4; ++t) hrow[t * 16 + lo] = acc[t][i];
        }
    }
}

// ---------------------------------------------------------------------------
// Kernel 2: per-node scores s_src = h.a[:64], s_dst = h.a[64:].
// One wave32 per node: each lane takes 2 columns, wave-shuffle reduction.
// ---------------------------------------------------------------------------
__global__ void __launch_bounds__(256)
score_kernel(const float* __restrict__ h, const float* __restrict__ a,
             float* __restrict__ s_src, float* __restrict__ s_dst) {
    int gid  = blockIdx.x * 256 + threadIdx.x;
    int node = gid >> 5;
    int lane = gid & 31;
    if (node >= NN) return;

    v2f hv = *(const v2f*)(h + (size_t)node * OUT_F + 2 * lane);
    float ps = hv.x * a[2 * lane]         + hv.y * a[2 * lane + 1];
    float pd = hv.x * a[OUT_F + 2 * lane] + hv.y * a[OUT_F + 2 * lane + 1];
#pragma unroll
    for (int off = 16; off > 0; off >>= 1) {
        ps += __shfl_down(ps, off, 32);
        pd += __shfl_down(pd, off, 32);
    }
    if (lane == 0) { s_src[node] = ps; s_dst[node] = pd; }
}

// ---------------------------------------------------------------------------
// Kernel 3: per-edge scatter. One wave32 per edge; lanes gather float2 of
// h[dst] (coalesced 256B), scale by edge_e, HW-atomic-add into hp[src].
// hp (25.6MB) + h (25.6MB) are L2-resident on the 192MB L2.
// ---------------------------------------------------------------------------
__global__ void __launch_bounds__(256)
edge_kernel(const int* __restrict__ edge, const float* __restrict__ h,
            const float* __restrict__ s_src, const float* __restrict__ s_dst,
            float* __restrict__ hp, float* __restrict__ rowsum) {
    long long gid = (long long)blockIdx.x * 256 + threadIdx.x;
    long long e   = gid >> 5;
    int lane      = (int)(gid & 31);
    if (e >= NE) return;

    // edge shape (2, NE): row 0 = src, row 1 = dst (harness delivers int32)
    int sidx = __builtin_amdgcn_readfirstlane(edge[e]);
    int didx = __builtin_amdgcn_readfirstlane(edge[NE + e]);

    float score = s_src[sidx] + s_dst[didx];
    float lr    = score > 0.0f ? score : ALPHA_NEG * score;
    float ee    = __expf(-lr);

    v2f hv = *(const v2f*)(h + (size_t)didx * OUT_F + 2 * lane);
    float* dst = hp + (size_t)sidx * OUT_F + 2 * lane;
    global_atomic_fadd(dst,     hv.x * ee);
    global_atomic_fadd(dst + 1, hv.y * ee);
    if (lane == 0) global_atomic_fadd(rowsum + sidx, ee);
}

// ---------------------------------------------------------------------------
// Kernel 4: h_prime /= rowsum; ELU. In place on d_out.
// ---------------------------------------------------------------------------
__global__ void __launch_bounds__(256)
finalize_kernel(float* __restrict__ out, const float* __restrict__ rowsum) {
    int idx = blockIdx.x * 256 + threadIdx.x;
    if (idx >= NN * OUT_F) return;
    float v = out[idx] / rowsum[idx >> 6];
    out[idx] = v > 0.0f ? v : (__expf(v) - 1.0f);
}

extern "C" void kernel_launch(void* const* d_in, const int* in_sizes, int n_in,
                              void* d_out, int out_size, void* d_ws, size_t ws_size,
                              hipStream_t stream) {
    const float* input = (const float*)d_in[0];
    const int*   edge  = (const int*)d_in[1];   // integer input -> int32 per harness
    const float* W     = (const float*)d_in[2];
    const float* a     = (const float*)d_in[3];
    float* out = (float*)d_out;

    char*  ws     = (char*)d_ws;
    float* h      = (float*)ws;                                  // NN*64 f32
    float* s_src  = (float*)(ws + (size_t)NN * OUT_F * sizeof(float));
    float* s_dst  = s_src + NN;
    float* rowsum = s_dst + NN;

    zero_kernel   <<<(NN * OUT_F + 255) / 256, 256, 0, stream>>>(out, rowsum);
    gemm_h_kernel <<<(NN + 127) / 128,          256, 0, stream>>>(input, W, h);
    score_kernel  <<<(NN * 32 + 255) / 256,     256, 0, stream>>>(h, a, s_src, s_dst);
    edge_kernel   <<<(int)(((long long)NE * 32 + 255) / 256), 256, 0, stream>>>(
                      edge, h, s_src, s_dst, out, rowsum);
    finalize_kernel<<<(NN * OUT_F + 255) / 256, 256, 0, stream>>>(out, rowsum);
}